// CRF_21191368638655
// MI455X (gfx1250) — compile-verified
//
#include <hip/hip_runtime.h>
#include <hip/hip_bf16.h>
#include <stdint.h>

#define B_ 64
#define T_ 512
#define H_ 768
#define L_ 9

typedef __attribute__((ext_vector_type(2))) float v2f;
typedef __attribute__((ext_vector_type(8))) float v8f;

// ---------------------------------------------------------------------------
// Kernel 1: logits[b,t,l] = sum_h x[b,t,h] * W[h,l] + bias[l]
// One wave32 per 16-row output tile, V_WMMA_F32_16X16X4_F32 over K=768.
// A (16x4 f32): lane L -> row m=L&15, K pair 2*(L>>4)   (2 VGPRs)
// B (4x16 f32): lane L -> col n=L&15, K pair 2*(L>>4)   (2 VGPRs)
// D (16x16 f32, v8f): lane L -> col n=L&15, rows 8*(L>>4)+r
// ---------------------------------------------------------------------------
__global__ __launch_bounds__(128) void crf_logits_wmma(
    const float* __restrict__ x, const float* __restrict__ W,
    const float* __restrict__ bias, float* __restrict__ logits)
{
  const int wave = (int)((blockIdx.x * blockDim.x + threadIdx.x) >> 5);
  const int lane = (int)(threadIdx.x & 31);
  const int m    = lane & 15;        // A row within tile
  const int hi   = lane >> 4;        // half-wave -> K pair select
  const int n    = m;                // B/D column
  const int ncl  = (n < L_) ? n : (L_ - 1);   // clamp address; junk cols discarded

  const float* xr = x + (size_t)(wave * 16 + m) * H_ + 2 * hi;
  const float* wr = W + (size_t)(2 * hi) * L_ + ncl;

  v8f acc = {};
#pragma unroll 8
  for (int kk = 0; kk < H_; kk += 4) {
    if ((kk & 127) == 0) __builtin_prefetch(xr + kk + 128, 0, 3);
    v2f a = *(const v2f*)(xr + kk);          // x[row, kk+2hi .. kk+2hi+1]
    v2f bm;
    bm[0] = wr[(size_t)kk * L_];             // W[kk+2hi,   ncl]
    bm[1] = wr[(size_t)(kk + 1) * L_];       // W[kk+2hi+1, ncl]
    acc = __builtin_amdgcn_wmma_f32_16x16x4_f32(
        false, a, false, bm, (short)0, acc, false, false);
  }

  if (n < L_) {
    const float bn = bias[n];
    float* out = logits + (size_t)(wave * 16 + hi * 8) * L_ + n;
#pragma unroll
    for (int r = 0; r < 8; ++r) out[(size_t)r * L_] = acc[r] + bn;
  }
}

// ---------------------------------------------------------------------------
// Kernel 2: per-batch CRF scan. One wave32 per batch element.
// The batch's 18KB logits slice is staged to LDS once via
// global_load_async_to_lds_b128 (ASYNCcnt), so the 512 serially-dependent
// scan steps read alpha inputs at LDS latency instead of L2 latency.
// Lane j (j<9) owns state j of both recursions; alpha broadcast via shuffles;
// backpointers in LDS; lane 0 backtracks; all lanes compute score/stats.
// ---------------------------------------------------------------------------
__global__ __launch_bounds__(32) void crf_scan(
    const float* __restrict__ logits, const float* __restrict__ trans,
    const int* __restrict__ label, const int* __restrict__ seqlen,
    float* __restrict__ vit_out, float* __restrict__ part)
{
  __shared__ float slg[T_ * L_];        // 18432 B: this batch's logits
  __shared__ unsigned char bp[T_][L_];  // backpointers
  __shared__ unsigned char tags[T_];    // decoded tags

  const int b    = blockIdx.x;
  const int lane = (int)threadIdx.x;
  const int j    = lane;
  const bool jv  = (j < L_);
  const int jc   = jv ? j : 0;
  const float NEG = -1.0e30f;

  const float* lg = logits + (size_t)b * T_ * L_;

  // ---- async stage: global logits slice -> LDS (36 x 32 lanes x 16B) ----
  {
    const unsigned ldsbase = (unsigned)(uintptr_t)(&slg[0]);
    const char* gbase = (const char*)lg;
#pragma unroll 4
    for (int it = 0; it < (T_ * L_ * 4) / (32 * 16); ++it) {
      const unsigned off  = (unsigned)(it * 32 + lane) * 16u;
      const unsigned ldsa = ldsbase + off;
      const char*    ga   = gbase + off;
      asm volatile("global_load_async_to_lds_b128 %0, %1, off"
                   :: "v"(ldsa), "v"(ga) : "memory");
    }
    asm volatile("s_wait_asynccnt 0" ::: "memory");
  }
  __syncthreads();

  float tcol[L_];                      // trans[i][j] for this lane's column j
#pragma unroll
  for (int i = 0; i < L_; ++i) tcol[i] = trans[i * L_ + jc];

  const int sl = seqlen[b];

  float aZ = jv ? slg[j] : NEG;        // logsumexp recursion
  float aV = aZ;                       // Viterbi (max) recursion

  for (int t = 1; t < T_; ++t) {
    const float logit = slg[t * L_ + jc];
    const bool valid = (t < sl);

    float scZ[L_], scV[L_];
#pragma unroll
    for (int i = 0; i < L_; ++i) {
      const float azi = __shfl(aZ, i, 32);
      const float avi = __shfl(aV, i, 32);
      scZ[i] = azi + tcol[i];
      scV[i] = avi + tcol[i];
    }
    // Viterbi max / first-argmax
    float mx = scV[0]; int am = 0;
#pragma unroll
    for (int i = 1; i < L_; ++i) if (scV[i] > mx) { mx = scV[i]; am = i; }
    // logsumexp
    float mz = scZ[0];
#pragma unroll
    for (int i = 1; i < L_; ++i) mz = fmaxf(mz, scZ[i]);
    float s = 0.f;
#pragma unroll
    for (int i = 0; i < L_; ++i) s += __expf(scZ[i] - mz);

    const float newZ = mz + __logf(s) + logit;
    const float newV = mx + logit;
    if (valid && jv) { aZ = newZ; aV = newV; }
    if (jv) bp[t][j] = (unsigned char)(valid ? am : j);
  }
  __syncthreads();

  // argmax_j aV (first-index wins on ties), all lanes converge on answer
  float best = aV; int bidx = jv ? j : (L_ - 1);
#pragma unroll
  for (int off = 16; off >= 1; off >>= 1) {
    const float ov = __shfl_xor(best, off, 32);
    const int   oi = __shfl_xor(bidx, off, 32);
    if (ov > best || (ov == best && oi < bidx)) { best = ov; bidx = oi; }
  }

  // logZ = logsumexp_j aZ
  float mzA = aZ;                      // NEG on inactive lanes
#pragma unroll
  for (int off = 16; off >= 1; off >>= 1) mzA = fmaxf(mzA, __shfl_xor(mzA, off, 32));
  float e = jv ? __expf(aZ - mzA) : 0.f;
#pragma unroll
  for (int off = 16; off >= 1; off >>= 1) e += __shfl_xor(e, off, 32);
  const float logZ = mzA + __logf(e);

  // backtrack (matches reference masking: invalid steps have identity bp)
  if (lane == 0) {
    int tag = bidx;
    tags[T_ - 1] = (unsigned char)tag;
    for (int t = T_ - 1; t >= 1; --t) {
      tag = bp[t][tag];
      tags[t - 1] = (unsigned char)tag;
    }
  }
  __syncthreads();

  // score + statistics, lanes strided over T (unary served from LDS)
  float unary = 0.f, tsum = 0.f, cor = 0.f, tpv = 0.f, tnv = 0.f, fpv = 0.f;
  for (int t = lane; t < T_; t += 32) {
    const int lab = label[b * T_ + t];
    const int vt  = (int)tags[t];
    vit_out[(size_t)b * T_ + t] = (float)vt;
    const bool mk = (t < sl);
    if (mk) unary += slg[t * L_ + lab];
    if (mk && t >= 1) tsum += trans[label[b * T_ + t - 1] * L_ + lab];
    if (mk && vt == lab) cor += 1.f;
    if (lab > 0 && vt == lab) tpv += 1.f;     // reference tp: no mask
    if (lab > 0 && vt != lab) tnv += 1.f;     // reference tn: no mask
    if (mk && lab == 0 && vt > 0) fpv += 1.f;
  }
#pragma unroll
  for (int off = 16; off >= 1; off >>= 1) {
    unary += __shfl_xor(unary, off, 32);
    tsum  += __shfl_xor(tsum,  off, 32);
    cor   += __shfl_xor(cor,   off, 32);
    tpv   += __shfl_xor(tpv,   off, 32);
    tnv   += __shfl_xor(tnv,   off, 32);
    fpv   += __shfl_xor(fpv,   off, 32);
  }
  if (lane == 0) {
    part[b]          = (unary + tsum) - logZ;   // log-likelihood
    part[B_ + b]     = cor;
    part[2 * B_ + b] = tpv;
    part[3 * B_ + b] = tnv;
    part[4 * B_ + b] = fpv;
  }
}

// ---------------------------------------------------------------------------
// Kernel 3: deterministic sequential reduction over B=64 batches.
// ---------------------------------------------------------------------------
__global__ void crf_finalize(const float* __restrict__ part,
                             const int* __restrict__ seqlen,
                             float* __restrict__ out5)
{
  if (threadIdx.x == 0 && blockIdx.x == 0) {
    float ll = 0.f, cor = 0.f, tp = 0.f, tn = 0.f, fp = 0.f, slsum = 0.f;
    for (int b = 0; b < B_; ++b) {
      ll  += part[b];
      cor += part[B_ + b];
      tp  += part[2 * B_ + b];
      tn  += part[3 * B_ + b];
      fp  += part[4 * B_ + b];
      slsum += (float)seqlen[b];
    }
    out5[0] = tp;
    out5[1] = tn;
    out5[2] = fp;
    out5[3] = -ll / (float)B_;     // loss = mean(-ll)
    out5[4] = cor / slsum;         // accuracy
  }
}

extern "C" void kernel_launch(void* const* d_in, const int* in_sizes, int n_in,
                              void* d_out, int out_size, void* d_ws, size_t ws_size,
                              hipStream_t stream) {
  (void)in_sizes; (void)n_in; (void)out_size; (void)ws_size;
  const float* x      = (const float*)d_in[0];
  const float* W      = (const float*)d_in[1];
  const float* bias   = (const float*)d_in[2];
  const float* trans  = (const float*)d_in[3];
  const int*   label  = (const int*)d_in[4];
  const int*   seqlen = (const int*)d_in[5];
  // d_in[6] = mask: unused (recomputed from seqlen)

  float* logits  = (float*)d_ws;                       // B*T*L floats (1.18 MB)
  float* part    = logits + (size_t)B_ * T_ * L_;      // 5*B floats
  float* vit_out = (float*)d_out;                      // B*T floats
  float* scalars = vit_out + (size_t)B_ * T_;          // 5 floats

  const int tiles = (B_ * T_) / 16;                    // 2048 wave-tiles
  crf_logits_wmma<<<dim3(tiles / 4), dim3(128), 0, stream>>>(x, W, bias, logits);
  crf_scan<<<dim3(B_), dim3(32), 0, stream>>>(logits, trans, label, seqlen,
                                              vit_out, part);
  crf_finalize<<<dim3(1), dim3(1), 0, stream>>>(part, seqlen, scalars);
}